// SpectralConv3d_RC_50577534877778
// MI455X (gfx1250) — compile-verified
//
#include <hip/hip_runtime.h>

// SpectralConv3d via truncated-mode separable DFTs.
// x:[4,32,64,64,64] f32, weights:[4,32,32,8,8,8,2] f32 -> out:[4,32,64,64,64] f32
//
// Stage 1 (WMMA): per (b,i,h) slab, A1[w][n] = X[w][t] * G1[t][n], n = [re kz0..7 | im kz0..7]
// Stage 2 (VALU): contract w with e^{-i}, Stage 3: contract h with e^{-i}
// Stage 4 (VALU): per-mode 32x32 complex channel mix with corner-selected weights
// Stage 5 (VALU): expand h with e^{+i}, Stage 6: expand w with e^{+i}
// Stage 7 (WMMA): out[w][t] = D6[w][k] * G7[k][t], G7 folds (kz?2:1)/64^3 weights.

typedef float v2f __attribute__((ext_vector_type(2)));
typedef float v8f __attribute__((ext_vector_type(8)));

#define TWO_PI_OVER_64 0.09817477042468103f

// workspace float offsets
#define OFF_G1   0                       // [t=64][n=16]        fwd-T table
#define OFF_FW2  1024                    // [ky=16][w=64][2]    fwd-W (cos,-sin)
#define OFF_FH3  3072                    // [kx=16][h=64][2]    fwd-H (cos,-sin)
#define OFF_IH5  5120                    // [h=64][kx=16][2]    inv-H (cos,+sin)
#define OFF_IW6  7168                    // [w=64][ky=16][2]    inv-W (cos,+sin)
#define OFF_G7   9216                    // [k=16][t=64]        inv-T table (weighted)
#define OFF_A1   16384                   // [slab=8192][w=64][n=16]  (reused as D6)
#define OFF_B2   (16384 + 8388608)       // [slab=8192][ky=16][kz=8][2] (reused as C5)
#define OFF_X3   (OFF_B2 + 2097152)      // [bi=128][kx=16][ky=16][kz=8][2]
#define OFF_Y4   (OFF_X3 + 1048576)      // [bo=128][kx=16][ky=16][kz=8][2]

__global__ void init_tables(float* __restrict__ ws) {
  for (int idx = threadIdx.x; idx < 10240; idx += blockDim.x) {
    float v = 0.0f;
    if (idx < 1024) {                       // G1: n<8 -> cos, n>=8 -> -sin  (e^{-i})
      int t = idx >> 4, n = idx & 15, kz = n & 7;
      float th = (float)((kz * t) & 63) * TWO_PI_OVER_64;
      v = (n < 8) ? cosf(th) : -sinf(th);
    } else if (idx < 3072) {                // FW2 (cos, -sin)
      int e = (idx - 1024) >> 1;
      int ky = e >> 6, w = e & 63;
      int kv = (ky < 8) ? ky : (48 + ky);
      float th = (float)((kv * w) & 63) * TWO_PI_OVER_64;
      v = (idx & 1) ? -sinf(th) : cosf(th);
    } else if (idx < 5120) {                // FH3 (cos, -sin)
      int e = (idx - 3072) >> 1;
      int kx = e >> 6, h = e & 63;
      int kv = (kx < 8) ? kx : (48 + kx);
      float th = (float)((kv * h) & 63) * TWO_PI_OVER_64;
      v = (idx & 1) ? -sinf(th) : cosf(th);
    } else if (idx < 7168) {                // IH5 [h][kx] (cos, +sin)
      int e = (idx - 5120) >> 1;
      int h = e >> 4, kx = e & 15;
      int kv = (kx < 8) ? kx : (48 + kx);
      float th = (float)((kv * h) & 63) * TWO_PI_OVER_64;
      v = (idx & 1) ? sinf(th) : cosf(th);
    } else if (idx < 9216) {                // IW6 [w][ky] (cos, +sin)
      int e = (idx - 7168) >> 1;
      int w = e >> 4, ky = e & 15;
      int kv = (ky < 8) ? ky : (48 + ky);
      float th = (float)((kv * w) & 63) * TWO_PI_OVER_64;
      v = (idx & 1) ? sinf(th) : cosf(th);
    } else {                                // G7: k<8 -> wt*cos, k>=8 -> -wt*sin
      int j = idx - 9216;
      int k = j >> 6, t = j & 63, kz = k & 7;
      float wt = ((kz == 0) ? 1.0f : 2.0f) * (1.0f / 262144.0f);
      float th = (float)((kz * t) & 63) * TWO_PI_OVER_64;
      v = (k < 8) ? wt * cosf(th) : -wt * sinf(th);
    }
    ws[idx] = v;
  }
}

// ---- Stage 1: forward T transform.  GEMM 64(w) x 64(t) x 16(n) per slab via WMMA.
__global__ void fwd_t_kernel(const float* __restrict__ x, float* __restrict__ A1,
                             const float* __restrict__ G1) {
  __shared__ float sG[1024];
  for (int i = threadIdx.x; i < 1024; i += 256) sG[i] = G1[i];
  __syncthreads();
  const int wave = blockIdx.x * 8 + (threadIdx.x >> 5);
  const int lane = threadIdx.x & 31;
  const int slab = wave >> 2;                 // (b*32+i)*64 + h
  const int w0   = (wave & 3) * 16;
  const int m    = lane & 15;
  const int koff = (lane < 16) ? 0 : 2;
  const float* __restrict__ Xrow = x + slab * 4096 + (w0 + m) * 64;
  v8f acc = {};
#pragma unroll
  for (int k0 = 0; k0 < 64; k0 += 4) {
    v2f a; a.x = Xrow[k0 + koff];            a.y = Xrow[k0 + koff + 1];
    v2f b; b.x = sG[(k0 + koff) * 16 + m];   b.y = sG[(k0 + koff + 1) * 16 + m];
    acc = __builtin_amdgcn_wmma_f32_16x16x4_f32(false, a, false, b,
                                                (short)0, acc, false, false);
  }
  float* __restrict__ out = A1 + slab * 1024;
  const int roff = (lane < 16) ? 0 : 8;
#pragma unroll
  for (int r = 0; r < 8; ++r) out[(w0 + r + roff) * 16 + m] = acc[r];
}

// ---- Stage 2: forward W transform (complex, K=64), one block per slab.
__global__ void fwd_w_kernel(const float* __restrict__ A1, float* __restrict__ B2,
                             const float* __restrict__ FW2) {
  __shared__ float sA[1024];
  const int slab = blockIdx.x;
  const float* __restrict__ src = A1 + slab * 1024;
  for (int i = threadIdx.x; i < 1024; i += 128) sA[i] = src[i];
  __syncthreads();
  const int ky = threadIdx.x >> 3, kz = threadIdx.x & 7;
  float re = 0.f, im = 0.f;
  for (int w = 0; w < 64; ++w) {
    float ar = sA[w * 16 + kz], ai = sA[w * 16 + 8 + kz];
    float twr = FW2[(ky * 64 + w) * 2], twi = FW2[(ky * 64 + w) * 2 + 1];
    re += twr * ar - twi * ai;
    im += twr * ai + twi * ar;
  }
  float* __restrict__ dst = B2 + ((slab * 16 + ky) * 8 + kz) * 2;
  dst[0] = re; dst[1] = im;
}

// ---- Stage 3: forward H transform (complex, K=64), thread per output mode.
__global__ void fwd_h_kernel(const float* __restrict__ B2, float* __restrict__ X3,
                             const float* __restrict__ FH3) {
  const int t  = blockIdx.x * 256 + threadIdx.x;     // 262144 total
  const int bi = t >> 11;
  const int md = t & 2047;
  const int kx = md >> 7, ky = (md >> 3) & 15, kz = md & 7;
  float re = 0.f, im = 0.f;
  for (int h = 0; h < 64; ++h) {
    const float* __restrict__ p = B2 + (((bi * 64 + h) * 16 + ky) * 8 + kz) * 2;
    float ar = p[0], ai = p[1];
    float twr = FH3[(kx * 64 + h) * 2], twi = FH3[(kx * 64 + h) * 2 + 1];
    re += twr * ar - twi * ai;
    im += twr * ai + twi * ar;
  }
  float* __restrict__ dst = X3 + (((bi * 16 + kx) * 16 + ky) * 8 + kz) * 2;
  dst[0] = re; dst[1] = im;
}

// ---- Stage 4: per-mode 32x32 complex channel mix. One wave per (b, mode), lane = out channel.
__global__ void mix_kernel(const float* __restrict__ X3, const float* __restrict__ W,
                           float* __restrict__ Y4) {
  const int wave = blockIdx.x * 8 + (threadIdx.x >> 5);   // 8192 waves
  const int lane = threadIdx.x & 31;                      // o
  const int b  = wave >> 11;
  const int md = wave & 2047;
  const int kx = md >> 7, ky = (md >> 3) & 15, kz = md & 7;
  const int wsel = ((kx >= 8) ? 1 : 0) + ((ky >= 8) ? 2 : 0);
  const int m1 = kx & 7, m2 = ky & 7, m3 = kz;
  float re = 0.f, im = 0.f;
  for (int i = 0; i < 32; ++i) {
    const float* __restrict__ xp =
        X3 + ((((b * 32 + i) * 16 + kx) * 16 + ky) * 8 + kz) * 2;
    float xr = xp[0], xi = xp[1];
    const float* __restrict__ wp =
        W + ((((((wsel * 32 + i) * 32 + lane) * 8 + m1) * 8 + m2) * 8 + m3) * 2);
    float wr = wp[0], wi = wp[1];
    re += xr * wr - xi * wi;
    im += xr * wi + xi * wr;
  }
  float* __restrict__ yp = Y4 + ((((b * 32 + lane) * 16 + kx) * 16 + ky) * 8 + kz) * 2;
  yp[0] = re; yp[1] = im;
}

// ---- Stage 5: inverse H expansion (complex, K=16), thread per output.
__global__ void inv_h_kernel(const float* __restrict__ Y4, float* __restrict__ C5,
                             const float* __restrict__ IH5) {
  const int t   = blockIdx.x * 256 + threadIdx.x;   // 1048576 total
  const int bo  = t >> 13;
  const int rem = t & 8191;
  const int h = rem >> 7, ky = (rem >> 3) & 15, kz = rem & 7;
  float re = 0.f, im = 0.f;
#pragma unroll
  for (int kx = 0; kx < 16; ++kx) {
    const float* __restrict__ yp = Y4 + (((bo * 16 + kx) * 16 + ky) * 8 + kz) * 2;
    float twr = IH5[(h * 16 + kx) * 2], twi = IH5[(h * 16 + kx) * 2 + 1];
    re += twr * yp[0] - twi * yp[1];
    im += twr * yp[1] + twi * yp[0];
  }
  float* __restrict__ dst = C5 + (((bo * 64 + h) * 16 + ky) * 8 + kz) * 2;
  dst[0] = re; dst[1] = im;
}

// ---- Stage 6: inverse W expansion (complex, K=16), block per (b,o,h) slab.
__global__ void inv_w_kernel(const float* __restrict__ C5, float* __restrict__ D6,
                             const float* __restrict__ IW6) {
  __shared__ float sC[256];
  const int slab = blockIdx.x;                     // 8192 blocks, 512 threads
  const float* __restrict__ src = C5 + slab * 256;
  if (threadIdx.x < 256) sC[threadIdx.x] = src[threadIdx.x];
  __syncthreads();
  const int w = threadIdx.x >> 3, kz = threadIdx.x & 7;
  float re = 0.f, im = 0.f;
#pragma unroll
  for (int ky = 0; ky < 16; ++ky) {
    float twr = IW6[(w * 16 + ky) * 2], twi = IW6[(w * 16 + ky) * 2 + 1];
    float cr = sC[(ky * 8 + kz) * 2], ci = sC[(ky * 8 + kz) * 2 + 1];
    re += twr * cr - twi * ci;
    im += twr * ci + twi * cr;
  }
  float* __restrict__ d = D6 + slab * 1024;        // [w][n], n = [re kz | im kz]
  d[w * 16 + kz]     = re;
  d[w * 16 + 8 + kz] = im;
}

// ---- Stage 7: inverse T to real output.  GEMM 64(w) x 16(k) x 64(t) per slab via WMMA.
__global__ void inv_t_kernel(const float* __restrict__ D6, float* __restrict__ out,
                             const float* __restrict__ G7) {
  __shared__ float sG[1024];
  for (int i = threadIdx.x; i < 1024; i += 256) sG[i] = G7[i];
  __syncthreads();
  const int wave = blockIdx.x * 8 + (threadIdx.x >> 5);
  const int lane = threadIdx.x & 31;
  const int slab = wave >> 2;                 // (b*32+o)*64 + h
  const int w0   = (wave & 3) * 16;
  const int m    = lane & 15;
  const int koff = (lane < 16) ? 0 : 2;
  const float* __restrict__ Drow = D6 + slab * 1024 + (w0 + m) * 16;
  v8f acc[4] = {{}, {}, {}, {}};
#pragma unroll
  for (int k0 = 0; k0 < 16; k0 += 4) {
    v2f a; a.x = Drow[k0 + koff]; a.y = Drow[k0 + koff + 1];
#pragma unroll
    for (int tt = 0; tt < 4; ++tt) {
      v2f b;
      b.x = sG[(k0 + koff) * 64 + tt * 16 + m];
      b.y = sG[(k0 + koff + 1) * 64 + tt * 16 + m];
      acc[tt] = __builtin_amdgcn_wmma_f32_16x16x4_f32(false, a, false, b,
                                                      (short)0, acc[tt], false, false);
    }
  }
  float* __restrict__ o = out + slab * 4096;
  const int roff = (lane < 16) ? 0 : 8;
#pragma unroll
  for (int tt = 0; tt < 4; ++tt)
#pragma unroll
    for (int r = 0; r < 8; ++r)
      o[(w0 + r + roff) * 64 + tt * 16 + m] = acc[tt][r];
}

extern "C" void kernel_launch(void* const* d_in, const int* in_sizes, int n_in,
                              void* d_out, int out_size, void* d_ws, size_t ws_size,
                              hipStream_t stream) {
  const float* x  = (const float*)d_in[0];   // [4,32,64,64,64]
  const float* wt = (const float*)d_in[1];   // [4,32,32,8,8,8,2]
  float* out = (float*)d_out;                // [4,32,64,64,64]
  float* ws  = (float*)d_ws;

  float* G1  = ws + OFF_G1;
  float* FW2 = ws + OFF_FW2;
  float* FH3 = ws + OFF_FH3;
  float* IH5 = ws + OFF_IH5;
  float* IW6 = ws + OFF_IW6;
  float* G7  = ws + OFF_G7;
  float* A1  = ws + OFF_A1;   // also D6
  float* B2  = ws + OFF_B2;   // also C5
  float* X3  = ws + OFF_X3;
  float* Y4  = ws + OFF_Y4;

  init_tables<<<1, 256, 0, stream>>>(ws);
  fwd_t_kernel<<<4096, 256, 0, stream>>>(x, A1, G1);     // 32768 waves
  fwd_w_kernel<<<8192, 128, 0, stream>>>(A1, B2, FW2);
  fwd_h_kernel<<<1024, 256, 0, stream>>>(B2, X3, FH3);
  mix_kernel  <<<1024, 256, 0, stream>>>(X3, wt, Y4);
  inv_h_kernel<<<4096, 256, 0, stream>>>(Y4, B2 /*C5*/, IH5);
  inv_w_kernel<<<8192, 512, 0, stream>>>(B2 /*C5*/, A1 /*D6*/, IW6);
  inv_t_kernel<<<4096, 256, 0, stream>>>(A1 /*D6*/, out, G7);
}